// PrototypicalNetwork_80874234183813
// MI455X (gfx1250) — compile-verified
//
#include <hip/hip_runtime.h>

typedef float v2f __attribute__((ext_vector_type(2)));
typedef float v8f __attribute__((ext_vector_type(8)));

#define BATCH      65536
#define DIM        512
#define NWAY       64
#define LDS_STRIDE (DIM + 4)   // pad so lane lm hits bank 4*lm: conflict-free ds_load_b64
#define MT         2           // 16-row M tiles per wave
#define NT         4           // 16-col N tiles (covers all 64 prototypes)
#define ROWS_PER_WAVE (16 * MT)
#define WAVES_PER_WG  8
#define ROWS_PER_WG   (ROWS_PER_WAVE * WAVES_PER_WG)   // 256

__global__ __launch_bounds__(256, 2)
void cosine_wmma_kernel(const float* __restrict__ x,
                        const float* __restrict__ proto,
                        float* __restrict__ out)
{
    __shared__ float ldsB[NWAY * LDS_STRIDE];   // 64 * 516 * 4B = 129 KB

    const int tid = threadIdx.x;

    // ---- stage prototypes (64 x 512 f32 = 128 KB) into padded LDS ----
    for (int i4 = tid; i4 < (NWAY * DIM) / 4; i4 += 256) {
        const int row = i4 >> 7;          // / (DIM/4)
        const int c4  = i4 & 127;
        const float4 v = ((const float4*)proto)[i4];
        *(float4*)&ldsB[row * LDS_STRIDE + (c4 << 2)] = v;
    }
    __syncthreads();

    const int wave = tid >> 5;
    const int lane = tid & 31;
    const int lh   = lane >> 4;    // lane half -> K pair {2lh, 2lh+1}
    const int lm   = lane & 15;

    const int waveRow = blockIdx.x * ROWS_PER_WG + wave * ROWS_PER_WAVE;

    v8f acc[MT][NT] = {};
    float xsq[MT] = {};
    float psq[NT] = {};

    // A fragment source: x[(waveRow + 16m + lm)][k + 2*lh .. +1]
    const float* xptr[MT];
    #pragma unroll
    for (int m = 0; m < MT; ++m)
        xptr[m] = x + (size_t)(waveRow + 16 * m + lm) * DIM + 2 * lh;

    // B fragment source (LDS): proto[(16n + lm)][k + 2*lh .. +1]
    int bidx[NT];
    #pragma unroll
    for (int n = 0; n < NT; ++n)
        bidx[n] = (16 * n + lm) * LDS_STRIDE + 2 * lh;

    for (int k = 0; k < DIM; k += 4) {
        v2f a[MT], b[NT];
        #pragma unroll
        for (int m = 0; m < MT; ++m)
            a[m] = *(const v2f*)(xptr[m] + k);
        #pragma unroll
        for (int n = 0; n < NT; ++n)
            b[n] = *(const v2f*)&ldsB[bidx[n] + k];

        // fused sum-of-squares partials (v_pk_fma_f32, co-executes with XDL WMMA)
        #pragma unroll
        for (int m = 0; m < MT; ++m)
            xsq[m] = fmaf(a[m].y, a[m].y, fmaf(a[m].x, a[m].x, xsq[m]));
        #pragma unroll
        for (int n = 0; n < NT; ++n)
            psq[n] = fmaf(b[n].y, b[n].y, fmaf(b[n].x, b[n].x, psq[n]));

        #pragma unroll
        for (int m = 0; m < MT; ++m)
            #pragma unroll
            for (int n = 0; n < NT; ++n)
                acc[m][n] = __builtin_amdgcn_wmma_f32_16x16x4_f32(
                    /*neg_a=*/false, a[m], /*neg_b=*/false, b[n],
                    /*c_mod=*/(short)0, acc[m][n],
                    /*reuse_a=*/false, /*reuse_b=*/false);
    }

    // combine the two lane-halves' K-subsets -> full sum of squares
    #pragma unroll
    for (int m = 0; m < MT; ++m) xsq[m] += __shfl_xor(xsq[m], 16, 32);
    #pragma unroll
    for (int n = 0; n < NT; ++n) psq[n] += __shfl_xor(psq[n], 16, 32);

    float pn[NT];
    #pragma unroll
    for (int n = 0; n < NT; ++n) pn[n] = sqrtf(psq[n]);  // norm of col 16n+lm (matches C/D lane)

    #pragma unroll
    for (int m = 0; m < MT; ++m) {
        const float xn_lane = sqrtf(xsq[m]);             // norm of row waveRow+16m+lm
        #pragma unroll
        for (int r = 0; r < 8; ++r) {
            const float xn  = __shfl(xn_lane, r + 8 * lh, 32);  // row r+8*lh of this tile
            const int   row = waveRow + 16 * m + r + 8 * lh;
            #pragma unroll
            for (int n = 0; n < NT; ++n) {
                const float denom = fmaxf(xn * pn[n], 1e-8f);
                // -1/denom via hardware v_rcp_f32 (~1 ulp); negate folds into the mul
                const float ninv = -__builtin_amdgcn_rcpf(denom);
                out[(size_t)row * NWAY + 16 * n + lm] = acc[m][n][r] * ninv;
            }
        }
    }
}

extern "C" void kernel_launch(void* const* d_in, const int* in_sizes, int n_in,
                              void* d_out, int out_size, void* d_ws, size_t ws_size,
                              hipStream_t stream) {
    const float* x     = (const float*)d_in[0];   // [65536, 512] f32
    const float* proto = (const float*)d_in[1];   // [64, 512] f32
    float* out         = (float*)d_out;           // [65536, 64] f32
    (void)in_sizes; (void)n_in; (void)out_size; (void)d_ws; (void)ws_size;

    dim3 grid(BATCH / ROWS_PER_WG);   // 256 workgroups
    dim3 block(256);                  // 8 wave32
    cosine_wmma_kernel<<<grid, block, 0, stream>>>(x, proto, out);
}